// CapsuleLayer_72421738545823
// MI455X (gfx1250) — compile-verified
//
#include <hip/hip_runtime.h>
#include <math.h>

// CapsNet dynamic routing, fully fused for MI455X (gfx1250).
//   x:  [B=256, R=1152, I=8]  f32
//   W:  [C=10,  R=1152, I=8, O=16] f32
//   out:[C, B, 1, O] f32  (40960 elements)
//
// One 256-thread workgroup (8 wave32) per (c,b) pair. priors[1152][16] f32
// lives in LDS (CDNA5: 320 KB per WGP). All 3 routing iterations run out of
// LDS; both routing reductions map onto V_WMMA_F32_16X16X4_F32 (exact f32):
//   s[o]     = sum_r probs[r]*pri[r,o]   (A = probs broadcast down rows)
//   dlogit[r]= sum_o pri[r,o]*out[o]     (B = out broadcast across cols)
// Trip counts are compile-time uniform (36 and 9 per wave) so EXEC stays
// all-ones through every WMMA — no lane-exit exec masking.

typedef __attribute__((ext_vector_type(2))) float v2f;
typedef __attribute__((ext_vector_type(8))) float v8f;

#define C_CAPS   10
#define B_SZ     256
#define R_SZ     1152
#define I_SZ     8
#define O_SZ     16
#define PSTRIDE  17          // pad priors rows: bank-conflict-free on 64 banks
#define NTHREADS 256
#define NWAVES   8
#define NITER    3

__device__ __forceinline__ float wave_max(float v) {
    #pragma unroll
    for (int off = 16; off > 0; off >>= 1)
        v = fmaxf(v, __shfl_xor(v, off, 32));
    return v;
}
__device__ __forceinline__ float wave_sum(float v) {
    #pragma unroll
    for (int off = 16; off > 0; off >>= 1)
        v += __shfl_xor(v, off, 32);
    return v;
}

__global__ __launch_bounds__(NTHREADS)
void capsule_routing_kernel(const float* __restrict__ x,
                            const float* __restrict__ W,
                            float* __restrict__ out)
{
    extern __shared__ float smem[];
    float* pri     = smem;                      // R_SZ * PSTRIDE
    float* logits  = pri + R_SZ * PSTRIDE;      // R_SZ
    float* probs   = logits + R_SZ;             // R_SZ   (unnormalized exp)
    float* s_arr   = probs + R_SZ;              // 16
    float* out_arr = s_arr + O_SZ;              // 16
    float* red     = out_arr + O_SZ;            // NWAVES

    const int tid  = threadIdx.x;
    const int lane = tid & 31;
    const int wave = __builtin_amdgcn_readfirstlane(tid >> 5);  // wave-uniform
    const int c    = blockIdx.x % C_CAPS;
    const int b    = blockIdx.x / C_CAPS;

    const float* xb = x + (size_t)b * R_SZ * I_SZ;
    const float* Wc = W + (size_t)c * R_SZ * I_SZ * O_SZ;

    // ---- priors[r][o] = sum_i x[b,r,i] * W[c,r,i,o]  (coalesced, 4 o's/thread)
    #pragma unroll 2
    for (int pass = 0; pass < (R_SZ * 4) / NTHREADS; ++pass) {
        int g  = pass * NTHREADS + tid;         // (r, o-group) index
        int r  = g >> 2;
        int o4 = (g & 3) * 4;
        const float4* xr = (const float4*)(xb + r * I_SZ);
        float4 x0 = xr[0], x1 = xr[1];
        float xi[8] = { x0.x, x0.y, x0.z, x0.w, x1.x, x1.y, x1.z, x1.w };
        const float* Wr = Wc + (size_t)r * I_SZ * O_SZ + o4;
        float4 acc = make_float4(0.f, 0.f, 0.f, 0.f);
        #pragma unroll
        for (int i = 0; i < I_SZ; ++i) {
            float4 w = *(const float4*)(Wr + i * O_SZ);
            acc.x += xi[i] * w.x;  acc.y += xi[i] * w.y;
            acc.z += xi[i] * w.z;  acc.w += xi[i] * w.w;
        }
        float* p = pri + r * PSTRIDE + o4;
        p[0] = acc.x; p[1] = acc.y; p[2] = acc.z; p[3] = acc.w;
    }
    for (int r = tid; r < R_SZ; r += NTHREADS) logits[r] = 0.f;
    __syncthreads();

    const int ksel = (lane < 16) ? 0 : 2;   // A/B K-pair held by this half-wave
    const int ncol = lane & 15;             // B/D column index
    float inv = 1.0f;

    for (int it = 0; it < NITER; ++it) {
        // ---- softmax over r: block max (wave butterfly + 8-entry LDS stage)
        float lmax = -INFINITY;
        for (int r = tid; r < R_SZ; r += NTHREADS) lmax = fmaxf(lmax, logits[r]);
        lmax = wave_max(lmax);
        if (lane == 0) red[wave] = lmax;
        __syncthreads();
        float mx = red[0];
        #pragma unroll
        for (int wv = 1; wv < NWAVES; ++wv) mx = fmaxf(mx, red[wv]);
        __syncthreads();

        // ---- exp + sum (probs kept unnormalized; fold 1/sum into s) ----
        float lsum = 0.f;
        for (int r = tid; r < R_SZ; r += NTHREADS) {
            float e = expf(logits[r] - mx);
            probs[r] = e; lsum += e;
        }
        lsum = wave_sum(lsum);
        if (lane == 0) red[wave] = lsum;
        if (tid < O_SZ) s_arr[tid] = 0.f;
        __syncthreads();
        float tot = red[0];
        #pragma unroll
        for (int wv = 1; wv < NWAVES; ++wv) tot += red[wv];
        inv = 1.0f / tot;

        // ---- s[o] = sum_r probs[r]*pri[r,o] via WMMA f32 16x16x4 ----
        // A[m,k] = probs[r0+k] (rows identical), B[k,n] = pri[r0+k][n]
        {
            v8f acc = {0.f,0.f,0.f,0.f,0.f,0.f,0.f,0.f};
            #pragma unroll 4
            for (int j = 0; j < (R_SZ / 4) / NWAVES; ++j) {     // 36, uniform
                int r0 = (j * NWAVES + wave) * 4;
                v2f A, Bv;
                A.x  = probs[r0 + ksel];
                A.y  = probs[r0 + ksel + 1];
                Bv.x = pri[(r0 + ksel)     * PSTRIDE + ncol];
                Bv.y = pri[(r0 + ksel + 1) * PSTRIDE + ncol];
                acc = __builtin_amdgcn_wmma_f32_16x16x4_f32(
                          false, A, false, Bv, (short)0, acc, false, false);
            }
            // every D row equal: row M=0 lives in acc[0] on lanes 0..15
            if (lane < 16) atomicAdd(&s_arr[lane], acc[0]);
        }
        __syncthreads();

        // ---- squash ----
        if (tid == 0) {
            float sv[O_SZ]; float sq = 0.f;
            #pragma unroll
            for (int o = 0; o < O_SZ; ++o) { sv[o] = s_arr[o] * inv; sq += sv[o] * sv[o]; }
            float scale = sqrtf(sq) / (1.0f + sq);
            #pragma unroll
            for (int o = 0; o < O_SZ; ++o) out_arr[o] = sv[o] * scale;
        }
        __syncthreads();

        // ---- logits[r] += sum_o pri[r,o]*out[o] via WMMA ----
        // A[m,k] = pri[r0+m][o0+k], B[k,n] = out[o0+k] (columns identical)
        if (it < NITER - 1) {
            const int m = lane & 15;
            for (int jc = 0; jc < (R_SZ / 16) / NWAVES; ++jc) { // 9, uniform
                int r0 = (jc * NWAVES + wave) * 16;
                v8f acc = {0.f,0.f,0.f,0.f,0.f,0.f,0.f,0.f};
                #pragma unroll
                for (int o0 = 0; o0 < O_SZ; o0 += 4) {
                    v2f A, Bv;
                    A.x  = pri[(r0 + m) * PSTRIDE + o0 + ksel];
                    A.y  = pri[(r0 + m) * PSTRIDE + o0 + ksel + 1];
                    Bv.x = out_arr[o0 + ksel];
                    Bv.y = out_arr[o0 + ksel + 1];
                    acc = __builtin_amdgcn_wmma_f32_16x16x4_f32(
                              false, A, false, Bv, (short)0, acc, false, false);
                }
                // D[m,n] identical over n; VGPR j holds M=j (lanes<16) / M=j+8
                if (lane == 0) {
                    #pragma unroll
                    for (int j = 0; j < 8; ++j) logits[r0 + j] += acc[j];
                } else if (lane == 16) {
                    #pragma unroll
                    for (int j = 0; j < 8; ++j) logits[r0 + 8 + j] += acc[j];
                }
            }
            __syncthreads();
        }
    }

    // out[c, b, 0, o]
    if (tid < O_SZ) out[((size_t)c * B_SZ + b) * O_SZ + tid] = out_arr[tid];
}

extern "C" void kernel_launch(void* const* d_in, const int* in_sizes, int n_in,
                              void* d_out, int out_size, void* d_ws, size_t ws_size,
                              hipStream_t stream) {
    (void)in_sizes; (void)n_in; (void)out_size; (void)d_ws; (void)ws_size;
    const float* x = (const float*)d_in[0];
    const float* W = (const float*)d_in[1];
    float* out = (float*)d_out;

    size_t lds_bytes = (size_t)(R_SZ * PSTRIDE + 2 * R_SZ + 2 * O_SZ + NWAVES) * sizeof(float);
    dim3 grid(C_CAPS * B_SZ);
    dim3 block(NTHREADS);
    capsule_routing_kernel<<<grid, block, lds_bytes, stream>>>(x, W, out);
}